// RepeatEncoder_317827580115
// MI455X (gfx1250) — compile-verified
//
#include <hip/hip_runtime.h>

// LIF spiking encoder: out[t][b][c][l] = spike_t( in[b][l][c] ), T=16 steps of
//   v = v + (x - v)/2 ; s = (v >= 1) ; v = 0 if s   (hard reset, x constant over t)
// Memory-bound: 16 MB in + 256 MB out ~= 11.7 us @ 23.3 TB/s. No matmul -> no WMMA.
// CDNA5 paths: async global->LDS tile staging (ASYNCcnt + s_wait_asynccnt) for the
// C<->L transpose, nontemporal 128-bit streaming stores, wave32 block sizing.
// v3: fuse recurrence with store-value generation (drop mask build + re-expand)
//     to cut VALU from ~482 to ~350/thread, keeping VALU under the HBM roofline.

#define T_STEPS 16
#define B_DIM   64
#define L_DIM   512
#define C_DIM   128

#define TILE      32
#define LDS_PITCH 33   // pad to avoid LDS bank conflicts on transposed reads

#if defined(__has_builtin)
#if __has_builtin(__builtin_amdgcn_global_load_async_to_lds_b32)
#define HAVE_ASYNC_LDS 1
#endif
#if __has_builtin(__builtin_amdgcn_s_wait_asynccnt)
#define HAVE_WAIT_ASYNC 1
#endif
#endif

typedef __attribute__((address_space(1))) int gint_t;   // global (prints as __device__ int*)
typedef __attribute__((address_space(3))) int lint_t;   // LDS
typedef float f32x4 __attribute__((ext_vector_type(4)));

__global__ __launch_bounds__(256) void lif_repeat_encoder(const float* __restrict__ in,
                                                          float* __restrict__ out) {
  __shared__ float tile[TILE * LDS_PITCH];

  const int l0 = blockIdx.x * TILE;   // L tile origin
  const int c0 = blockIdx.y * TILE;   // C tile origin
  const int b  = blockIdx.z;

  const int tid = threadIdx.x;

  // ---- Phase 1: stage 32x32 (l, c) tile into LDS, coalesced along C ----
  {
    const int cx  = tid & 31;  // contiguous c within a wave -> coalesced global reads
    const int ly0 = tid >> 5;  // 8 rows per pass
    const float* src = in + ((size_t)b * L_DIM + (size_t)l0) * C_DIM + c0;
#pragma unroll
    for (int p = 0; p < 4; ++p) {
      const int l = ly0 + p * 8;
      const float* g = src + (size_t)l * C_DIM + cx;
      float* s = &tile[l * LDS_PITCH + cx];
#if defined(HAVE_ASYNC_LDS)
      __builtin_amdgcn_global_load_async_to_lds_b32((gint_t*)g, (lint_t*)s, 0, 0);
#else
      *s = __builtin_nontemporal_load(g);
#endif
    }
  }

#if defined(HAVE_ASYNC_LDS)
#if defined(HAVE_WAIT_ASYNC)
  __builtin_amdgcn_s_wait_asynccnt(0);
#else
  asm volatile("s_wait_asynccnt 0" ::: "memory");
#endif
#endif
  __syncthreads();

  // ---- Phase 2: transposed fused compute + streaming stores, coalesced along L ----
  const int k   = tid & 7;    // 0..7  -> group of 4 consecutive l
  const int cy  = tid >> 3;   // 0..31 -> c row within tile
  const int lx0 = k * 4;

  float x[4], v[4];
#pragma unroll
  for (int j = 0; j < 4; ++j) {
    x[j] = tile[(lx0 + j) * LDS_PITCH + cy];
    v[j] = 0.0f;
  }

  const size_t tstride = (size_t)B_DIM * C_DIM * L_DIM;  // elements per t-plane
  const size_t obase =
      (((size_t)b * C_DIM) + (size_t)(c0 + cy)) * L_DIM + (size_t)(l0 + lx0);
  float* oplane = out + obase;

#pragma unroll
  for (int t = 0; t < T_STEPS; ++t) {
    f32x4 vv;
#pragma unroll
    for (int j = 0; j < 4; ++j) {
      v[j] = v[j] + (x[j] - v[j]) * 0.5f;       // sub + fma
      const bool sp = (v[j] >= 1.0f);           // v_cmp
      vv[j] = sp ? 1.0f : 0.0f;                 // cndmask -> store operand directly
      v[j]  = sp ? 0.0f : v[j];                 // cndmask (hard reset)
    }
    // output plane never re-read and exceeds L2 -> non-temporal streaming store
    __builtin_nontemporal_store(vv, (f32x4*)(oplane + (size_t)t * tstride));
  }
}

extern "C" void kernel_launch(void* const* d_in, const int* in_sizes, int n_in,
                              void* d_out, int out_size, void* d_ws, size_t ws_size,
                              hipStream_t stream) {
  (void)in_sizes; (void)n_in; (void)out_size; (void)d_ws; (void)ws_size;
  const float* in = (const float*)d_in[0];
  float* out = (float*)d_out;
  dim3 grid(L_DIM / TILE, C_DIM / TILE, B_DIM);  // (16, 4, 64)
  lif_repeat_encoder<<<grid, 256, 0, stream>>>(in, out);
}